// CatEncoderCross_15951508537718
// MI455X (gfx1250) — compile-verified
//
#include <hip/hip_runtime.h>

// ---------------------------------------------------------------------------
// Problem: out[b,n,m,o] = (in1[b,n,:] @ W[:768,o]) + (in2[b,m,:] @ W[768:,o]) + bias[o]
//   in1, in2: (4,128,768) f32   W: (1536,768) f32   bias: (768,) f32
//   out: (4,128,128,768) f32  -> 201 MB streaming write  (bandwidth bound)
// Strategy:
//   Phase 1: WMMA f32 GEMMs -> P1 (=in1@W1 + bias), P2 (=in2@W2) in workspace (3 MB, L2 resident)
//   Phase 2: streaming broadcast add, non-temporal float4 stores of the 201 MB output
// ---------------------------------------------------------------------------

typedef float v2f __attribute__((ext_vector_type(2)));
typedef float v4f __attribute__((ext_vector_type(4)));
typedef float v8f __attribute__((ext_vector_type(8)));

#define DMODEL 768
#define NROWS  512              // 4 * 128 rows per input
#define TILES_N (DMODEL / 16)   // 48
#define TILES_M (NROWS / 16)    // 32
#define TILES_PER_MAT (TILES_M * TILES_N)  // 1536

// One wave32 computes one 16x16 tile of P = X @ Wsub (K = 768, stepped by 4).
__global__ __launch_bounds__(256) void cat_gemm_wmma_f32(
    const float* __restrict__ X1,   // (512, 768)
    const float* __restrict__ X2,   // (512, 768)
    const float* __restrict__ W,    // (1536, 768) row-major
    const float* __restrict__ bias, // (768,)
    float* __restrict__ P1,         // (512, 768)
    float* __restrict__ P2)         // (512, 768)
{
    const int wave = blockIdx.x * (blockDim.x >> 5) + (threadIdx.x >> 5);
    const int lane = threadIdx.x & 31;
    const int half = lane >> 4;   // 0: K={0,1}, 1: K={2,3} sub-slab of each x4 step
    const int lsub = lane & 15;   // M index (A) / N index (B, C/D)

    const int mat = wave / TILES_PER_MAT;      // 0 -> (X1, W1), 1 -> (X2, W2)
    const int t   = wave % TILES_PER_MAT;
    const int tM  = t / TILES_N;
    const int tN  = t % TILES_N;

    const float* __restrict__ X  = (mat == 0) ? X1 : X2;
    const float* __restrict__ Wb = W + (size_t)mat * DMODEL * DMODEL;
    float* __restrict__ P        = (mat == 0) ? P1 : P2;

    // A fragment source: row (tM*16 + lsub), contiguous K pairs (8B aligned).
    const float* __restrict__ Arow = X + (size_t)(tM * 16 + lsub) * DMODEL;
    // B fragment source: column (tN*16 + lsub), stride DMODEL between K's.
    const float* __restrict__ Bcol = Wb + (tN * 16 + lsub);

    v8f acc = {};
    for (int kk = 0; kk < DMODEL; kk += 4) {
        const int k0 = kk + 2 * half;
        v2f a = *(const v2f*)(Arow + k0);              // A: K=k0, k0+1
        v2f b;
        b.x = Bcol[(size_t)k0 * DMODEL];               // B: K=k0
        b.y = Bcol[(size_t)(k0 + 1) * DMODEL];         // B: K=k0+1
        // (neg_a, A, neg_b, B, c_mod, C, reuse_a, reuse_b)
        acc = __builtin_amdgcn_wmma_f32_16x16x4_f32(
            false, a, false, b, (short)0, acc, false, false);
    }

    // C/D layout: VGPR v -> (M = half*8 + v, N = lsub)
    const int col = tN * 16 + lsub;
    const float bv = (mat == 0) ? bias[col] : 0.0f;    // fold bias into P1 only
#pragma unroll
    for (int v = 0; v < 8; ++v) {
        const int row = tM * 16 + half * 8 + v;
        P[(size_t)row * DMODEL + col] = acc[v] + bv;
    }
}

// Pure streaming kernel: out[((b*128+n)*128+m)] row of 768 = P1row + P2row.
// One float4 per thread; P1/P2 reads hit L2 (3 MB resident); output uses
// non-temporal stores so the 201 MB write stream does not evict P1/P2.
#define O4 (DMODEL / 4)   // 192 float4 per output row
__global__ __launch_bounds__(256) void cat_bcast_add(
    const float* __restrict__ P1,
    const float* __restrict__ P2,
    float* __restrict__ out)
{
    const int i = blockIdx.x * 256 + threadIdx.x;      // [0, 12582912), exact
    const int o4   = i % O4;
    int rest       = i / O4;
    const int m    = rest & 127;
    rest         >>= 7;
    const int n    = rest & 127;
    const int b    = rest >> 7;

    const v4f* __restrict__ p1 = (const v4f*)P1;
    const v4f* __restrict__ p2 = (const v4f*)P2;

    const v4f a = p1[(b * 128 + n) * O4 + o4];
    const v4f c = p2[(b * 128 + m) * O4 + o4];
    const v4f r = a + c;
    __builtin_nontemporal_store(r, (v4f*)out + i);
}

extern "C" void kernel_launch(void* const* d_in, const int* in_sizes, int n_in,
                              void* d_out, int out_size, void* d_ws, size_t ws_size,
                              hipStream_t stream) {
    const float* input1 = (const float*)d_in[0];   // (4,128,768)
    const float* input2 = (const float*)d_in[1];   // (4,128,768)
    const float* W      = (const float*)d_in[2];   // (1536,768)
    const float* bias   = (const float*)d_in[3];   // (768,)

    float* P1 = (float*)d_ws;                      // 512*768 f32 = 1.5 MB
    float* P2 = P1 + (size_t)NROWS * DMODEL;       // +1.5 MB

    // Phase 1: 3072 tiles, 8 waves (one tile each) per 256-thread block.
    const int total_waves = 2 * TILES_PER_MAT;     // 3072
    cat_gemm_wmma_f32<<<total_waves / 8, 256, 0, stream>>>(
        input1, input2, W, bias, P1, P2);

    // Phase 2: 4*128*128*192 float4 = 12,582,912 -> 49152 blocks of 256.
    const int total4 = 4 * 128 * 128 * O4;
    cat_bcast_add<<<total4 / 256, 256, 0, stream>>>(P1, P2, (float*)d_out);
}